// MultiBoxLoss_43147241456333
// MI455X (gfx1250) — compile-verified
//
#include <hip/hip_runtime.h>
#include <hip/hip_bf16.h>

// ---------------------------------------------------------------------------
// MultiBox (SSD) loss for MI455X / gfx1250.
// Memory-bound (~150 MB traffic -> ~6.5us @ 23.3 TB/s). No GEMM -> no WMMA.
// CDNA5 features used: async global->LDS staging (ASYNCcnt), 320KB LDS for
// per-sample in-LDS radix select (hard-negative mining), wave32 ballot scan.
// ---------------------------------------------------------------------------

#define NS   128
#define NB   8732
#define NL   21
#define K1T  256
#define NBLK ((NB + K1T - 1) / K1T)   // 35 blocks per sample in pass 1

#if defined(__has_builtin)
#  if __has_builtin(__builtin_amdgcn_global_load_async_to_lds_b128) && \
      __has_builtin(__builtin_amdgcn_s_wait_asynccnt)
#    define USE_ASYNC_LDS 1
#  endif
#endif
#ifndef USE_ASYNC_LDS
#  define USE_ASYNC_LDS 0
#endif

typedef int v4i __attribute__((ext_vector_type(4)));

// ---------------------------------------------------------------------------
// Pass 1: per-anchor log-softmax stats + positive box loss partials.
// One block handles 256 anchors of one sample. Predictions (84B/anchor) are
// staged into LDS with coalesced b128 async copies, then consumed stride-21
// (gcd(21,64)=1 -> bank-conflict free).
// ---------------------------------------------------------------------------
__global__ __launch_bounds__(K1T) void mb_pass1(
    const float* __restrict__ boxes, const float* __restrict__ preds,
    const float* __restrict__ tboxes, const int* __restrict__ targets,
    unsigned int* __restrict__ bg_bits, float* __restrict__ tlp,
    int* __restrict__ npos_part, float* __restrict__ box_part)
{
    __shared__ __align__(16) float sh_pred[K1T * NL];
    __shared__ float sh_f[K1T];
    __shared__ int   sh_i[K1T];

    const int tid = threadIdx.x;
    const int bx  = blockIdx.x;
    const int s   = blockIdx.y;
    const int a0  = bx * K1T;
    const int n_anch = min(K1T, NB - a0);
    const int nflt   = n_anch * NL;
    const int nvec4  = nflt >> 2;           // float offset is 16B aligned (proof: 21*(s*8732+256*bx) % 4 == 0)
    const float* gbase = preds + ((size_t)s * NB + a0) * NL;

#if USE_ASYNC_LDS
    for (int i = tid; i < nvec4; i += K1T) {
        __builtin_amdgcn_global_load_async_to_lds_b128(
            (v4i*)(gbase + i * 4), (v4i*)(sh_pred + i * 4), 0, 0);
    }
    for (int i = (nvec4 << 2) + tid; i < nflt; i += K1T) sh_pred[i] = gbase[i];
    __builtin_amdgcn_s_wait_asynccnt(0);
    __syncthreads();
#else
    for (int i = tid; i < nvec4; i += K1T)
        *(float4*)(sh_pred + i * 4) = *(const float4*)(gbase + i * 4);
    for (int i = (nvec4 << 2) + tid; i < nflt; i += K1T) sh_pred[i] = gbase[i];
    __syncthreads();
#endif

    float my_box = 0.0f;
    int   my_pos = 0;
    if (tid < n_anch) {
        const size_t gi = (size_t)s * NB + (a0 + tid);
        const float* p  = sh_pred + tid * NL;
        float m = p[0];
        #pragma unroll
        for (int j = 1; j < NL; ++j) m = fmaxf(m, p[j]);
        float sum = 0.0f;
        #pragma unroll
        for (int j = 0; j < NL; ++j) sum += __expf(p[j] - m);
        const float lse = m + __logf(sum);

        const int  tg  = targets[gi];
        const bool pos = tg > 0;
        const float bg = pos ? 1.0f : __expf(p[0] - lse);   // positives pushed to top
        bg_bits[gi] = __float_as_uint(bg);                  // positive floats: bit order == value order
        tlp[gi]     = p[tg] - lse;

        if (pos) {
            my_pos = 1;
            const float4 b = ((const float4*)boxes)[gi];
            const float4 t = ((const float4*)tboxes)[gi];
            float d, a;
            d = b.x - t.x; a = fabsf(d); my_box += (a < 1.0f) ? 0.5f * d * d : a - 0.5f;
            d = b.y - t.y; a = fabsf(d); my_box += (a < 1.0f) ? 0.5f * d * d : a - 0.5f;
            d = b.z - t.z; a = fabsf(d); my_box += (a < 1.0f) ? 0.5f * d * d : a - 0.5f;
            d = b.w - t.w; a = fabsf(d); my_box += (a < 1.0f) ? 0.5f * d * d : a - 0.5f;
        }
    }

    sh_f[tid] = my_box; sh_i[tid] = my_pos;
    __syncthreads();
    for (int off = K1T / 2; off > 0; off >>= 1) {
        if (tid < off) { sh_f[tid] += sh_f[tid + off]; sh_i[tid] += sh_i[tid + off]; }
        __syncthreads();
    }
    if (tid == 0) {
        box_part[s * NBLK + bx]  = sh_f[0];
        npos_part[s * NBLK + bx] = sh_i[0];
    }
}

// ---------------------------------------------------------------------------
// Pass 2: one block per sample. In-LDS radix select of the retain-th smallest
// bg bit-pattern (4 x 8-bit MSB-first passes), exact stable tie-break by
// anchor index via wave32 ballot prefix + cross-chunk running count, then
// masked cross-entropy accumulation (deterministic tree reduce).
// ---------------------------------------------------------------------------
__global__ __launch_bounds__(1024) void mb_select(
    const unsigned int* __restrict__ bg_bits, const float* __restrict__ tlp,
    const int* __restrict__ targets, const int* __restrict__ npos_part,
    float* __restrict__ dsum_s, int* __restrict__ dcnt_s, int* __restrict__ npos_s)
{
    __shared__ unsigned int sh_bits[NB];      // 34928 B: whole sample on-chip
    __shared__ unsigned int sh_hist[256];
    __shared__ unsigned int sh_red[1024];
    __shared__ float        sh_fred[1024];
    __shared__ unsigned int sh_scal[8];

    const int tid = threadIdx.x;
    const int s   = blockIdx.x;
    const int NT  = 1024;

    if (tid == 0) {
        int np = 0;
        for (int j = 0; j < NBLK; ++j) np += npos_part[s * NBLK + j];
        int ret = np * 3;                     // (npos*3.0f).astype(int32) == 3*npos exactly
        if (ret < 1)  ret = 1;
        if (ret > NB) ret = NB;
        sh_scal[4] = (unsigned)ret;
        sh_scal[5] = (unsigned)np;
    }
    for (int i = tid; i < NB; i += NT) sh_bits[i] = bg_bits[(size_t)s * NB + i];
    __syncthreads();

    const unsigned retain = sh_scal[4];
    unsigned k = retain - 1;                  // 0-based rank of threshold
    unsigned prefix = 0, pmask = 0;
    for (int p = 3; p >= 0; --p) {
        if (tid < 256) sh_hist[tid] = 0;
        __syncthreads();
        for (int i = tid; i < NB; i += NT) {
            unsigned v = sh_bits[i];
            if ((v & pmask) == prefix) atomicAdd(&sh_hist[(v >> (8 * p)) & 255u], 1u);
        }
        __syncthreads();
        if (tid == 0) {
            unsigned cum = 0, d = 255;
            for (unsigned j = 0; j < 256; ++j) {
                unsigned h = sh_hist[j];
                if (cum + h > k) { d = j; break; }
                cum += h;
            }
            sh_scal[0] = d; sh_scal[1] = k - cum;
        }
        __syncthreads();
        prefix |= sh_scal[0] << (8 * p);
        pmask  |= 0xFFu << (8 * p);
        k = sh_scal[1];
        __syncthreads();
    }
    const unsigned T = prefix;                // value at rank retain-1

    unsigned cl = 0;
    for (int i = tid; i < NB; i += NT) cl += (sh_bits[i] < T) ? 1u : 0u;
    sh_red[tid] = cl;
    __syncthreads();
    for (int off = 512; off > 0; off >>= 1) {
        if (tid < off) sh_red[tid] += sh_red[tid + off];
        __syncthreads();
    }
    const unsigned count_less = sh_red[0];
    const unsigned need = retain - count_less;   // #ties (==T) to keep, lowest indices first
    __syncthreads();
    if (tid == 0) sh_scal[3] = 0;                // running tie count across chunks
    __syncthreads();

    const int lane = tid & 31, wv = tid >> 5;
    float dsum = 0.0f; unsigned dcnt = 0;
    const int*   tg = targets + (size_t)s * NB;
    const float* tl = tlp     + (size_t)s * NB;

    for (int base = 0; base < NB; base += NT) {
        const int  i     = base + tid;
        const bool valid = i < NB;
        const unsigned bits = valid ? sh_bits[i] : 0xFFFFFFFFu;
        const unsigned tie  = (valid && bits == T) ? 1u : 0u;

        const unsigned long long bal = __ballot(tie != 0);   // wave32 mask
        const unsigned excl_in_wave = (unsigned)__popcll(bal & ((1ull << lane) - 1ull));
        if (lane == 0) sh_red[wv] = (unsigned)__popcll(bal);
        __syncthreads();
        if (tid == 0) {
            unsigned r = 0;
            for (int j = 0; j < 32; ++j) { unsigned t2 = sh_red[j]; sh_red[j] = r; r += t2; }
            sh_scal[2] = r;                                  // ties in this chunk
        }
        __syncthreads();
        const unsigned excl = sh_red[wv] + excl_in_wave;
        const unsigned run  = sh_scal[3];
        const bool inc_bg = valid && ((bits < T) || (tie && (run + excl) < need));
        if (valid) {
            const bool m = (tg[i] > 0) || inc_bg;
            if (m) { dsum -= tl[i]; dcnt++; }
        }
        __syncthreads();
        if (tid == 0) sh_scal[3] = run + sh_scal[2];
        __syncthreads();
    }

    sh_fred[tid] = dsum; sh_red[tid] = dcnt;
    __syncthreads();
    for (int off = 512; off > 0; off >>= 1) {
        if (tid < off) { sh_fred[tid] += sh_fred[tid + off]; sh_red[tid] += sh_red[tid + off]; }
        __syncthreads();
    }
    if (tid == 0) {
        dsum_s[s] = sh_fred[0];
        dcnt_s[s] = (int)sh_red[0];
        npos_s[s] = (int)sh_scal[5];
    }
}

// ---------------------------------------------------------------------------
// Pass 3: deterministic final reduction -> scalar loss.
// ---------------------------------------------------------------------------
__global__ __launch_bounds__(256) void mb_final(
    const float* __restrict__ dsum_s, const int* __restrict__ dcnt_s,
    const int* __restrict__ npos_s, const float* __restrict__ box_part,
    float* __restrict__ out)
{
    __shared__ float sf[256]; __shared__ int si[256];
    __shared__ int   sn[256]; __shared__ float sb[256];
    const int tid = threadIdx.x;
    float bx = 0.0f;
    for (int i = tid; i < NS * NBLK; i += 256) bx += box_part[i];
    float ds = 0.0f; int dc = 0, np = 0;
    if (tid < NS) { ds = dsum_s[tid]; dc = dcnt_s[tid]; np = npos_s[tid]; }
    sf[tid] = ds; si[tid] = dc; sn[tid] = np; sb[tid] = bx;
    __syncthreads();
    for (int off = 128; off > 0; off >>= 1) {
        if (tid < off) {
            sf[tid] += sf[tid + off]; si[tid] += si[tid + off];
            sn[tid] += sn[tid + off]; sb[tid] += sb[tid + off];
        }
        __syncthreads();
    }
    if (tid == 0) {
        const float det = sf[0] / (float)max(si[0], 1);
        const float box = sb[0] / (float)max(sn[0] * 4, 1);
        out[0] = det + box;
    }
}

// ---------------------------------------------------------------------------
extern "C" void kernel_launch(void* const* d_in, const int* in_sizes, int n_in,
                              void* d_out, int out_size, void* d_ws, size_t ws_size,
                              hipStream_t stream) {
    const float* boxes   = (const float*)d_in[0];   // [128,8732,4]
    const float* preds   = (const float*)d_in[1];   // [128,8732,21]
    const float* tboxes  = (const float*)d_in[2];   // [128,8732,4]
    const int*   targets = (const int*)d_in[3];     // [128,8732]
    float* out = (float*)d_out;

    char* ws = (char*)d_ws;
    const size_t SZ_ANCH = (size_t)NS * NB * 4;     // 4,470,784 B (16B aligned)
    unsigned int* bg_bits   = (unsigned int*)(ws);
    float*        tlp       = (float*)(ws + SZ_ANCH);
    int*          npos_part = (int*)(ws + 2 * SZ_ANCH);
    float*        box_part  = (float*)(ws + 2 * SZ_ANCH + (size_t)NS * NBLK * 4);
    float*        dsum_s    = (float*)(ws + 2 * SZ_ANCH + 2 * (size_t)NS * NBLK * 4);
    int*          dcnt_s    = (int*)  ((char*)dsum_s + 512);
    int*          npos_s    = (int*)  ((char*)dsum_s + 1024);

    dim3 g1(NBLK, NS, 1);
    mb_pass1<<<g1, K1T, 0, stream>>>(boxes, preds, tboxes, targets,
                                     bg_bits, tlp, npos_part, box_part);
    mb_select<<<NS, 1024, 0, stream>>>(bg_bits, tlp, targets, npos_part,
                                       dsum_s, dcnt_s, npos_s);
    mb_final<<<1, 256, 0, stream>>>(dsum_s, dcnt_s, npos_s, box_part, out);
}